// FALSECriterion_72516227826000
// MI455X (gfx1250) — compile-verified
//
#include <hip/hip_runtime.h>
#include <math.h>

typedef float v2f __attribute__((ext_vector_type(2)));
typedef float v8f __attribute__((ext_vector_type(8)));

#define BSZ   4096
#define HALF  2048
#define DDIM  128
#define INV_T 2.0f      // 1/T, T = 0.5
#define TVAL  0.5f
#define CWT   0.5f
#define PTH   0.8f

// Workspace layout (floats):
//   [0      , 4096 )  possim[i]  = exp(dot(e_i, e_pos)/T)
//   [4096   , 8192 )  negtotal[i]
//   [8192   , 12288)  S[i]
//   ws[12288] (as uint) gmin bits

// ---------------- Kernel 0: positive-pair sims + gmin init ----------------
__global__ __launch_bounds__(256) void k_posdot(const float* __restrict__ E,
                                                float* __restrict__ ws) {
    int i = blockIdx.x * blockDim.x + threadIdx.x;
    if (i == 0) {
        ((unsigned*)ws)[3 * BSZ] = 0x7F800000u;  // +inf
    }
    if (i < BSZ) {
        int p = (i + HALF) & (BSZ - 1);
        const float4* a = (const float4*)(E + (size_t)i * DDIM);
        const float4* b = (const float4*)(E + (size_t)p * DDIM);
        float acc = 0.0f;
#pragma unroll
        for (int k = 0; k < DDIM / 4; ++k) {
            float4 x = a[k], y = b[k];
            acc += x.x * y.x + x.y * y.y + x.z * y.z + x.w * y.w;
        }
        ws[i] = expf(acc * INV_T);
    }
}

// ---------------- Shared WMMA tile pipeline (pass A / pass B) ----------------
// Each wave owns rows [rowBase, rowBase+16) and sweeps all 256 column tiles.
// A-fragment (f32 16x4): lane l16 = M, K = 2*h + v (consecutive pair per lane).
// B-fragment (f32 4x16): lane l16 = N, K = 2*h + v (symmetric gather from E).
// Column tiles are register double-buffered (ping-pong b0/b1) so global-load
// latency overlaps the WMMA chains; two accumulators halve the RAW chain.
template <bool PASSB>
__device__ __forceinline__ void tile_pass(const float* __restrict__ E,
                                          float* __restrict__ ws) {
    const int lane    = threadIdx.x & 31;
    const int wave    = threadIdx.x >> 5;
    const int l16     = lane & 15;
    const int h       = lane >> 4;
    const int rowBase = blockIdx.x * 128 + wave * 16;

    // Load A fragments once (32 K-steps of 4): 64 VGPRs, reused for 256 tiles.
    v2f a[32];
    {
        const float* arow = E + (size_t)(rowBase + l16) * DDIM + 2 * h;
#pragma unroll
        for (int kk = 0; kk < 32; ++kk) a[kk] = *(const v2f*)(arow + kk * 4);
    }

    // Per-lane row metadata: this lane's C values cover rows rowBase + r + 8*h.
    float psim[8];
    bool  conf[8];
#pragma unroll
    for (int r = 0; r < 8; ++r) {
        psim[r] = ws[rowBase + r + 8 * h];
        conf[r] = PASSB ? (logf(psim[r]) * TVAL >= PTH) : false;
    }

    float gmin = 0.0f;
    if (PASSB) gmin = __uint_as_float(((const unsigned*)ws)[3 * BSZ]);

    float acc[8];
#pragma unroll
    for (int r = 0; r < 8; ++r) acc[r] = 0.0f;
    float dmin = 3.0f;  // filler value participates in the reference min

    v2f b0[32], b1[32];

    auto loadB = [&](v2f* b, int ct) {
        const float* brow = E + (size_t)(ct * 16 + l16) * DDIM + 2 * h;
#pragma unroll
        for (int kk = 0; kk < 32; ++kk) b[kk] = *(const v2f*)(brow + kk * 4);
    };

    auto compute = [&](const v2f* b, int ct) {
        // Two accumulator chains (even/odd K-steps) -> 2x WMMA ILP.
        v8f cA = {}, cB = {};
#pragma unroll
        for (int kk = 0; kk < 32; kk += 2) {
            cA = __builtin_amdgcn_wmma_f32_16x16x4_f32(
                     false, a[kk + 0], false, b[kk + 0], (short)0, cA, false, false);
            cB = __builtin_amdgcn_wmma_f32_16x16x4_f32(
                     false, a[kk + 1], false, b[kk + 1], (short)0, cB, false, false);
        }
        const v8f c = cA + cB;

        const int j = ct * 16 + l16;  // column index for this lane
#pragma unroll
        for (int r = 0; r < 8; ++r) {
            const int   i   = rowBase + r + 8 * h;
            const int   p   = (i + HALF) & (BSZ - 1);
            const float sim = expf(c[r] * INV_T);
            const bool validUpper = (j > i) && !((i < HALF) && (j == i + HALF));
            if (!PASSB) {
                if ((j != i) && (j != p)) acc[r] += sim;  // negtotal
                if (validUpper) dmin = fminf(dmin, fabsf(sim - psim[r]));
            } else {
                const float d = validUpper ? fabsf(sim - psim[r]) : 3.0f;
                if (conf[r] && (d == gmin) && (j != i) && (j != p))
                    acc[r] += sim;  // S_i
            }
        }
    };

    // Software-pipelined sweep over 256 column tiles (ping-pong registers).
    loadB(b0, 0);
    for (int ct = 0; ct < BSZ / 16; ct += 2) {
        loadB(b1, ct + 1);                          // in flight during compute(b0)
        compute(b0, ct);
        int ctn = ct + 2;
        if (ctn > BSZ / 16 - 1) ctn = BSZ / 16 - 1; // clamped redundant prefetch
        loadB(b0, ctn);                             // in flight during compute(b1)
        compute(b1, ct + 1);
    }

    // Deterministic per-row sum: reduce across the 16 lanes of each half-wave.
#pragma unroll
    for (int r = 0; r < 8; ++r) {
        float v = acc[r];
        v += __shfl_xor(v, 8, 16);
        v += __shfl_xor(v, 4, 16);
        v += __shfl_xor(v, 2, 16);
        v += __shfl_xor(v, 1, 16);
        if (l16 == 0) {
            float* out = ws + (PASSB ? 2 * BSZ : BSZ);
            out[rowBase + r + 8 * h] = v;  // exclusive row ownership -> plain store
        }
    }

    if (!PASSB) {
        float v = dmin;
        v = fminf(v, __shfl_xor(v, 16, 32));
        v = fminf(v, __shfl_xor(v, 8, 32));
        v = fminf(v, __shfl_xor(v, 4, 32));
        v = fminf(v, __shfl_xor(v, 2, 32));
        v = fminf(v, __shfl_xor(v, 1, 32));
        if (lane == 0) {
            // all diffs >= 0 -> uint bit order == float order; min is
            // order-independent so the atomic stays deterministic.
            atomicMin((unsigned*)ws + 3 * BSZ, __float_as_uint(v));
        }
    }
}

__global__ __launch_bounds__(256) void k_passA(const float* __restrict__ E,
                                               float* __restrict__ ws) {
    tile_pass<false>(E, ws);
}

__global__ __launch_bounds__(256) void k_passB(const float* __restrict__ E,
                                               float* __restrict__ ws) {
    tile_pass<true>(E, ws);
}

// ---------------- Kernel 3: final loss reduction ----------------
__global__ __launch_bounds__(256) void k_loss(const float* __restrict__ ws,
                                              float* __restrict__ out) {
    __shared__ float red[256];
    float s = 0.0f;
    for (int i = threadIdx.x; i < BSZ; i += 256) {
        const float psim = ws[i];
        const float negt = ws[BSZ + i];
        const float S    = ws[2 * BSZ + i];
        const float pos  = psim + CWT * S;
        const float neg  = negt - CWT * S;
        s += logf(pos / (pos + neg));
    }
    red[threadIdx.x] = s;
    __syncthreads();
    for (int st = 128; st > 0; st >>= 1) {
        if (threadIdx.x < st) red[threadIdx.x] += red[threadIdx.x + st];
        __syncthreads();
    }
    if (threadIdx.x == 0) out[0] = -red[0] / (float)BSZ;
}

extern "C" void kernel_launch(void* const* d_in, const int* in_sizes, int n_in,
                              void* d_out, int out_size, void* d_ws, size_t ws_size,
                              hipStream_t stream) {
    (void)in_sizes; (void)n_in; (void)out_size; (void)ws_size;
    const float* E   = (const float*)d_in[0];  // [4096,128] f32, L2-normalized
    float*       ws  = (float*)d_ws;           // needs 12289 floats (~48 KB)
    float*       out = (float*)d_out;          // scalar f32 loss

    k_posdot<<<BSZ / 256, 256, 0, stream>>>(E, ws);   // possim + gmin init
    k_passA <<<BSZ / 128, 256, 0, stream>>>(E, ws);   // negtotal + global min
    k_passB <<<BSZ / 128, 256, 0, stream>>>(E, ws);   // S_i (argmin selection)
    k_loss  <<<1,         256, 0, stream>>>(ws, out); // -mean(log(pos/(pos+neg)))
}